// PolicyNetwork_19129784336473
// MI455X (gfx1250) — compile-verified
//
#include <hip/hip_runtime.h>

// u[b,t] = w[t]*q0[b] + bias[t]   (B=32768, T=4096, f32 out: 512 MiB stores)
// Store-bandwidth bound (floor ~23us @ 23.3 TB/s). Compute on the WMMA pipe:
// D(16x16) = A(16x4)*B(4x16) + C with K=1 real column (q0) / row (w), bias in C.
// T is a compile-time constant in the fast path so all 64 stores per wave fold
// their offsets into the 24-bit instruction immediate off a single base.

typedef __attribute__((ext_vector_type(2))) float v2f;
typedef __attribute__((ext_vector_type(8))) float v8f;

#define BT_PER_WAVE 8   // 16-row b-tiles handled per wave (one t-tile fixed)

template <int T_CONST>
__global__ __launch_bounds__(256)
void affine_outer_wmma(const float* __restrict__ q0,
                       const float* __restrict__ w,
                       const float* __restrict__ bias,
                       float* __restrict__ out,
                       int T_rt, int tTiles, int totalWaves)
{
    const int T = (T_CONST > 0) ? T_CONST : T_rt;

    const int lane        = threadIdx.x & 31;
    const int waveInBlock = threadIdx.x >> 5;
    const int W           = blockIdx.x * 8 + waveInBlock;
    if (W >= totalWaves) return;          // uniform per wave; EXEC stays all-1s

    const int tTile = W % tTiles;
    const int bChnk = W / tTiles;
    const int t0    = tTile * 16;
    const int n     = lane & 15;          // output column within tile
    const bool lo   = lane < 16;

    // w-tile -> B matrix (4x16): only row K=0 nonzero.
    // B layout: VGPR0 lanes0-15 = row K=0, lanes16-31 = row K=2; VGPR1 = K=1/K=3.
    const float wv = w[t0 + n];
    const float bv = bias[t0 + n];
    v2f bmat;
    bmat.x = lo ? wv : 0.0f;
    bmat.y = 0.0f;

    // bias broadcast into C: every C VGPR holds bias[t0 + N] (N = lane%16).
    v8f c;
#pragma unroll
    for (int i = 0; i < 8; ++i) c[i] = bv;

    const int rowBase = bChnk * (BT_PER_WAVE * 16);   // first b-row of this wave
    const int mBase   = (lane >> 4) << 3;             // 0 (lanes 0-15) or 8

    // Single per-lane base index; all store offsets below are constants.
    const unsigned baseIdx = (unsigned)(rowBase + mBase) * (unsigned)T
                           + (unsigned)(t0 + n);
    const float* q0w = q0 + rowBase + n;

#pragma unroll
    for (int r = 0; r < BT_PER_WAVE; ++r) {
        // q0-tile -> A matrix (16x4): only column K=0 nonzero.
        // A layout: VGPR0 lanes0-15 = K=0 (M=lane), lanes16-31 = K=2; VGPR1 = K=1/3.
        const float qv = q0w[r * 16];     // constant immediate offset
        v2f a;
        a.x = lo ? qv : 0.0f;
        a.y = 0.0f;

        // D = A*B + C  =>  D[m][n] = q0[.]*w[.] + bias[.]  (exact in f32)
        v8f d = __builtin_amdgcn_wmma_f32_16x16x4_f32(
            /*neg_a=*/false, a, /*neg_b=*/false, bmat,
            /*c_mod=*/(short)0, c, /*reuse_a=*/false, /*reuse_b=*/false);

        // D VGPR i, lane L -> element (M = i + mBase, N = n): 16-lane contiguous
        // runs => coalesced. Write-once output > L2: non-temporal stores.
        // Offset (r*16 + i)*T is compile-time in the fast path (max ~1.9 MB < 2^23).
#pragma unroll
        for (int i = 0; i < 8; ++i) {
            __builtin_nontemporal_store(
                d[i], out + (baseIdx + (unsigned)((r * 16 + i) * T)));
        }
    }
}

extern "C" void kernel_launch(void* const* d_in, const int* in_sizes, int n_in,
                              void* d_out, int out_size, void* d_ws, size_t ws_size,
                              hipStream_t stream)
{
    const float* q0   = (const float*)d_in[0];  // [B]
    const float* w    = (const float*)d_in[1];  // [T]
    const float* bias = (const float*)d_in[2];  // [T]
    float* out        = (float*)d_out;          // [B, T]

    const int B = in_sizes[0];
    const int T = in_sizes[1];

    const int tTiles = T / 16;                                    // 256
    const int bTiles = B / 16;                                    // 2048
    const int chunks = (bTiles + BT_PER_WAVE - 1) / BT_PER_WAVE;  // 256
    const int totalWaves = tTiles * chunks;                       // 65536
    const int blocks = (totalWaves + 7) / 8;                      // 8192 x 8 waves

    if (T == 4096 && (B % (16 * BT_PER_WAVE)) == 0) {
        hipLaunchKernelGGL((affine_outer_wmma<4096>), dim3(blocks), dim3(256), 0,
                           stream, q0, w, bias, out, T, tTiles, totalWaves);
    } else {
        hipLaunchKernelGGL((affine_outer_wmma<0>), dim3(blocks), dim3(256), 0,
                           stream, q0, w, bias, out, T, tTiles, totalWaves);
    }
}